// _GCNDenseDilatedKnnGraph_42554535969007
// MI455X (gfx1250) — compile-verified
//
#include <hip/hip_runtime.h>
#include <hip/hip_bf16.h>

// ---------------------------------------------------------------------------
// Dilated KNN graph: B=2, C=64, N=8192, K=9, DILATION=2 (keep top-18, take ::2)
// Output: int32 edge_index [2, B, N, 9]  (neighbors, then center indices)
// ---------------------------------------------------------------------------

typedef float v2f __attribute__((ext_vector_type(2)));
typedef float v8f __attribute__((ext_vector_type(8)));

namespace {
constexpr int B   = 2;
constexpr int C   = 64;
constexpr int N   = 8192;
constexpr int KO  = 9;     // kept neighbors
constexpr int KT  = 18;    // K * DILATION tracked during selection
constexpr int WPB = 4;     // waves per block in the main kernel
}

// ---------------------------------------------------------------------------
// Kernel 1: L2-normalize along C and transpose [B,C,N] -> P[B,N,C]; sq = sum P^2
// ---------------------------------------------------------------------------
__global__ void __launch_bounds__(256)
knn_normalize_kernel(const float* __restrict__ x,
                     float* __restrict__ P,
                     float* __restrict__ sq)
{
    const int gid = blockIdx.x * blockDim.x + threadIdx.x;   // 0 .. B*N-1
    if (gid >= B * N) return;
    const int b  = gid / N;
    const int n  = gid % N;
    const float* xb = x + (size_t)b * C * N;

    float v[C];
    float s = 0.0f;
#pragma unroll
    for (int c = 0; c < C; ++c) {
        float t = xb[(size_t)c * N + n];      // coalesced across threads per c
        v[c] = t;
        s += t * t;
    }
    const float nrm = sqrtf(s);
    const float inv = 1.0f / fmaxf(nrm, 1e-12f);

    float s2 = 0.0f;
#pragma unroll
    for (int c = 0; c < C; ++c) {
        float t = v[c] * inv;
        v[c] = t;
        s2 += t * t;
    }
    float4* dst = (float4*)(P + (size_t)gid * C);
#pragma unroll
    for (int c = 0; c < C / 4; ++c) {
        dst[c] = make_float4(v[4 * c], v[4 * c + 1], v[4 * c + 2], v[4 * c + 3]);
    }
    sq[gid] = s2;
}

// ---------------------------------------------------------------------------
// Kernel 2: fused Gram-matrix (f32 WMMA) + per-row running top-18 selection.
// One wave owns 32 rows (two 16-row M tiles). A fragments live in registers
// for the whole kernel; B panels stream through L2. 16x16 dist tiles are
// staged in LDS; each lane maintains a sorted top-18 list for one row.
// ---------------------------------------------------------------------------
__global__ void __launch_bounds__(WPB * 32, 1)
knn_topk_wmma_kernel(const float* __restrict__ P,
                     const float* __restrict__ sq,
                     int* __restrict__ out)
{
    __shared__ float tile[WPB][32][17];   // [wave][row][col] (+1 pad)

    const int lane  = threadIdx.x & 31;
    const int w     = threadIdx.x >> 5;
    const int gwave = blockIdx.x * WPB + w;

    constexpr int wavesPerBatch = N / 32;            // 256
    const int b  = gwave / wavesPerBatch;
    const int r0 = (gwave % wavesPerBatch) * 32;

    const float* Pb  = P  + (size_t)b * N * C;
    const float* sqb = sq + (size_t)b * N;

    const int lo = lane & 15;
    const int hi = lane >> 4;

    // --- A fragments for the wave's 32 rows: 16 K-steps x 2 M-tiles ---------
    v2f A0[16], A1[16];
#pragma unroll
    for (int kk = 0; kk < 16; ++kk) {
        const int c0 = 4 * kk + 2 * hi;
        A0[kk] = *(const v2f*)(Pb + (size_t)(r0      + lo) * C + c0);
        A1[kk] = *(const v2f*)(Pb + (size_t)(r0 + 16 + lo) * C + c0);
    }

    // --- per-lane row-norms matching the accumulator layout -----------------
    float sqr0[8], sqr1[8];
#pragma unroll
    for (int j = 0; j < 8; ++j) {
        sqr0[j] = sqb[r0      + j + 8 * hi];
        sqr1[j] = sqb[r0 + 16 + j + 8 * hi];
    }

    // --- per-lane sorted top-18 (ascending distance) ------------------------
    float vals[KT];
    int   idxs[KT];
#pragma unroll
    for (int i = 0; i < KT; ++i) { vals[i] = 3.0e38f; idxs[i] = 0; }

    for (int ct = 0; ct < N / 16; ++ct) {
        const int cbase = ct * 16;

        // hint next column panel into cache (speculative, drop-on-fault)
        __builtin_prefetch(Pb + (size_t)(cbase + 16 + lo) * C, 0, 0);

        // B fragments: same layout as A fragments, built from the 16 columns
        v2f Bf[16];
#pragma unroll
        for (int kk = 0; kk < 16; ++kk) {
            const int c0 = 4 * kk + 2 * hi;
            Bf[kk] = *(const v2f*)(Pb + (size_t)(cbase + lo) * C + c0);
        }

        v8f acc0 = {};
        v8f acc1 = {};
#pragma unroll
        for (int kk = 0; kk < 16; ++kk) {
            acc0 = __builtin_amdgcn_wmma_f32_16x16x4_f32(
                       false, A0[kk], false, Bf[kk], (short)0, acc0, false, false);
            acc1 = __builtin_amdgcn_wmma_f32_16x16x4_f32(
                       false, A1[kk], false, Bf[kk], (short)0, acc1, false, false);
        }

        const float sqc = sqb[cbase + lo];

        // dist = sq_row - 2*dot + sq_col, staged to LDS as [32 rows][16 cols]
#pragma unroll
        for (int j = 0; j < 8; ++j) {
            tile[w][j + 8 * hi][lo]      = sqr0[j] + sqc - 2.0f * acc0[j];
            tile[w][16 + j + 8 * hi][lo] = sqr1[j] + sqc - 2.0f * acc1[j];
        }
        __syncthreads();

        // lane l scans its own row's 16 candidates (ascending index order ->
        // strict '<' insertion reproduces top_k's lower-index-first ties)
#pragma unroll 4
        for (int c = 0; c < 16; ++c) {
            const float d  = tile[w][lane][c];
            const int   id = cbase + c;
            if (d < vals[KT - 1]) {
#pragma unroll
                for (int i = KT - 1; i >= 0; --i) {
                    if (d < vals[i]) {
                        if (i < KT - 1) { vals[i + 1] = vals[i]; idxs[i + 1] = idxs[i]; }
                        vals[i] = d; idxs[i] = id;
                    }
                }
            }
        }
        __syncthreads();
    }

    // --- emit edge_index[:, :, :, ::2] --------------------------------------
    const int myrow = r0 + lane;
    int* out_nn = out + (((size_t)0 * B + b) * N + myrow) * KO;
    int* out_ct = out + (((size_t)1 * B + b) * N + myrow) * KO;
#pragma unroll
    for (int j = 0; j < KO; ++j) {
        out_nn[j] = idxs[2 * j];
        out_ct[j] = myrow;
    }
}

// ---------------------------------------------------------------------------
extern "C" void kernel_launch(void* const* d_in, const int* in_sizes, int n_in,
                              void* d_out, int out_size, void* d_ws, size_t ws_size,
                              hipStream_t stream)
{
    (void)in_sizes; (void)n_in; (void)out_size; (void)ws_size;

    const float* x = (const float*)d_in[0];
    float* P   = (float*)d_ws;                       // B*N*C floats (4 MB)
    float* sqv = P + (size_t)B * N * C;              // B*N floats
    int*   out = (int*)d_out;

    {
        const int threads = 256;
        const int blocks  = (B * N + threads - 1) / threads;   // 64
        knn_normalize_kernel<<<blocks, threads, 0, stream>>>(x, P, sqv);
    }
    {
        const int totalWaves = B * (N / 32);                   // 512
        const int blocks     = totalWaves / WPB;               // 128
        knn_topk_wmma_kernel<<<blocks, WPB * 32, 0, stream>>>(P, sqv, out);
    }
}